// NeuroMIRT_41901700939904
// MI455X (gfx1250) — compile-verified
//
#include <hip/hip_runtime.h>
#include <hip/hip_bf16.h>
#include <math.h>

// Problem constants (match reference setup_inputs).
#define BB 128
#define SS 200
#define DD 128
#define KK 4
#define SM1 199          // S-1 scan steps
#define EPS 1e-8f

typedef __attribute__((ext_vector_type(2))) float v2f;
typedef __attribute__((ext_vector_type(8))) float v8f;

// ---------------------------------------------------------------------------
// WMMA: D = A(16x4 f32) * B(4x16 f32) + C(16x16 f32), wave32.
// A layout (ISA 7.12.2): lanes 0-15 hold M=lane {K=k0,k0+1}; lanes 16-31 hold
// M=lane-16 {K=k0+2,k0+3}.  B symmetric: lane half selects K pair, lane%16 = N.
// C/D layout: VGPR r -> M=r (lanes 0-15) / M=r+8 (lanes 16-31), lane%16 = N.
// ---------------------------------------------------------------------------
__device__ __forceinline__ v8f wmma_f32_k4(v2f a, v2f b, v8f c) {
    return __builtin_amdgcn_wmma_f32_16x16x4_f32(
        /*neg_a=*/false, a, /*neg_b=*/false, b,
        /*c_mod=*/(short)0, c, /*reuse_a=*/false, /*reuse_b=*/false);
}

__device__ __forceinline__ float sigmoidf_(float x) {
    return 1.0f / (1.0f + expf(-x));
}

// Branch-free GEMM segment: acc += Arow[k0..k0+len) x Bmat[brow0+k .. ][n]
// Arow is a contiguous source row for this K segment.
__device__ __forceinline__ v8f gemm_seg(
    v8f acc, const float* __restrict__ Arow, const float* __restrict__ Bmat,
    int brow0, int len, int kh, int n) {
    for (int k0 = 0; k0 < len; k0 += 4) {
        const int ka = k0 + kh;
        const v2f a = *(const v2f*)(Arow + ka);          // global_load_b64
        v2f b;
        b.x = Bmat[(size_t)(brow0 + ka) * DD + n];
        b.y = Bmat[(size_t)(brow0 + ka + 1) * DD + n];
        acc = wmma_f32_k4(a, b, acc);
    }
    return acc;
}

// Segment where the A value is a per-row broadcast constant (no loads).
__device__ __forceinline__ v8f gemm_seg_const(
    v8f acc, float aval, const float* __restrict__ Bmat,
    int brow0, int len, int kh, int n) {
    v2f a; a.x = aval; a.y = aval;
    for (int k0 = 0; k0 < len; k0 += 4) {
        const int ka = k0 + kh;
        v2f b;
        b.x = Bmat[(size_t)(brow0 + ka) * DD + n];
        b.y = Bmat[(size_t)(brow0 + ka + 1) * DD + n];
        acc = wmma_f32_k4(a, b, acc);
    }
    return acc;
}

// ===========================================================================
// Kernel 1: inter[t] = concat(qe_t, ce_t) @ W_fuse + b_fuse          [B,D]
// grid = SM1 * 8 (t, m-tile), block = 256 (8 waves -> 8 n-tiles)
// ===========================================================================
__global__ __launch_bounds__(256) void k_fuse(
    const int* __restrict__ qseq, const int* __restrict__ cseq,
    const float* __restrict__ eq, const float* __restrict__ Wf,
    const float* __restrict__ bf, float* __restrict__ inter) {
    const int t    = blockIdx.x >> 3;
    const int mt   = blockIdx.x & 7;
    const int lane = threadIdx.x & 31;
    const int nt   = threadIdx.x >> 5;
    const int l16  = lane & 15;
    const int kh   = (lane >> 4) << 1;   // 0 or 2
    const int mh   = (lane >> 4) << 3;   // 0 or 8 (C/D rows)
    const int m    = mt * 16 + l16;
    const int n    = nt * 16 + l16;

    const int   qi   = qseq[m * SS + t];
    const float corr = (float)cseq[m * SS + t];
    const float* qrow = eq + (size_t)qi * DD;

    v8f acc = {};
    acc = gemm_seg      (acc, qrow, Wf, /*brow0=*/0,  DD, kh, n);  // qe part
    acc = gemm_seg_const(acc, corr, Wf, /*brow0=*/DD, DD, kh, n);  // ce part

    const float bias = bf[n];
    float* dst = inter + (size_t)t * BB * DD;
    #pragma unroll
    for (int r = 0; r < 8; ++r) {
        const int mm = mt * 16 + r + mh;
        dst[mm * DD + n] = acc[r] + bias;
    }
}

// ===========================================================================
// Kernel 2: absorb[t] = concat(inter, ute, hae) @ W_absorb + b_absorb
//           fpart[t]  = concat(ite, inter) @ W_forget[128:384] + b_forget
// grid = SM1 * 8, block = 256
// ===========================================================================
__global__ __launch_bounds__(256) void k_absorb_forget(
    const int* __restrict__ useq, const int* __restrict__ hseq,
    const int* __restrict__ iseq,
    const float* __restrict__ inter, const float* __restrict__ eut,
    const float* __restrict__ eha, const float* __restrict__ eit,
    const float* __restrict__ Wa, const float* __restrict__ ba,
    const float* __restrict__ Wg, const float* __restrict__ bg,
    float* __restrict__ absorb, float* __restrict__ fpart) {
    const int t    = blockIdx.x >> 3;
    const int mt   = blockIdx.x & 7;
    const int lane = threadIdx.x & 31;
    const int nt   = threadIdx.x >> 5;
    const int l16  = lane & 15;
    const int kh   = (lane >> 4) << 1;
    const int mh   = (lane >> 4) << 3;
    const int m    = mt * 16 + l16;
    const int n    = nt * 16 + l16;

    const float* interRow = inter + (size_t)t * BB * DD + (size_t)m * DD;
    const float* urow = eut + (size_t)useq[m * SS + t] * DD;
    const float* hrow = eha + (size_t)hseq[m * SS + t] * DD;
    const float* irow = eit + (size_t)iseq[m * SS + t] * DD;

    // ---- absorb: K = 384 = [inter | ute | hae] ----
    v8f accA = {};
    accA = gemm_seg(accA, interRow, Wa, /*brow0=*/0,      DD, kh, n);
    accA = gemm_seg(accA, urow,     Wa, /*brow0=*/DD,     DD, kh, n);
    accA = gemm_seg(accA, hrow,     Wa, /*brow0=*/2 * DD, DD, kh, n);
    {
        const float bias = ba[n];
        float* dst = absorb + (size_t)t * BB * DD;
        #pragma unroll
        for (int r = 0; r < 8; ++r) {
            const int mm = mt * 16 + r + mh;
            dst[mm * DD + n] = accA[r] + bias;
        }
    }

    // ---- fpart: K = 256 = [ite | inter], W_forget rows 128..383 ----
    v8f accF = {};
    accF = gemm_seg(accF, irow,     Wg, /*brow0=*/DD,     DD, kh, n);
    accF = gemm_seg(accF, interRow, Wg, /*brow0=*/2 * DD, DD, kh, n);
    {
        const float bias = bg[n];
        float* dst = fpart + (size_t)t * BB * DD;
        #pragma unroll
        for (int r = 0; r < 8; ++r) {
            const int mm = mt * 16 + r + mh;
            dst[mm * DD + n] = accF[r] + bias;
        }
    }
}

// ===========================================================================
// Kernel 3: per-(t,b) scalars.
//   qd1   = 10*sigmoid(disc[q_{t+1}])
//   qdiff_k = (cos(qe_t, cp_t[k]) + 1)/2
//   wk[t,b,k] = 0.5*qd1*m_k        (weight on cos(lc, cp_{t+1}[k]))
//   c1[t,b]   = qd1 * sum_k m_k*(0.5 - qdiff_k)
// so logit_t = sum_k cos(lc, cp1_k)*wk + c1. Also zero y[:, S-1].
// ===========================================================================
__global__ __launch_bounds__(256) void k_scalars(
    const int* __restrict__ qseq, const int* __restrict__ q2c,
    const int* __restrict__ q2cm,
    const float* __restrict__ eq, const float* __restrict__ eqd,
    const float* __restrict__ ec,
    float* __restrict__ wk, float* __restrict__ c1, float* __restrict__ out) {
    const int idx = blockIdx.x * blockDim.x + threadIdx.x;
    if (idx >= SM1 * BB) return;
    const int t = idx / BB;
    const int b = idx % BB;

    const int q0 = qseq[b * SS + t];
    const int q1 = qseq[b * SS + t + 1];
    const float qd1 = 10.0f * sigmoidf_(eqd[q1]);
    const float* qrow = eq + (size_t)q0 * DD;

    float na = 0.0f;
    for (int d = 0; d < DD; ++d) na += qrow[d] * qrow[d];
    const float inv_na = 1.0f / fmaxf(sqrtf(na), EPS);

    float c1v = 0.0f;
    #pragma unroll
    for (int k = 0; k < KK; ++k) {
        const int   cidx = q2c[q0 * KK + k];
        const float mk   = (float)q2cm[q0 * KK + k];
        const float* crow = ec + (size_t)cidx * DD;
        float dot = 0.0f, nb = 0.0f;
        for (int d = 0; d < DD; ++d) {
            dot += qrow[d] * crow[d];
            nb  += crow[d] * crow[d];
        }
        const float cosv  = dot * inv_na / fmaxf(sqrtf(nb), EPS);
        const float qdiff = (cosv + 1.0f) * 0.5f;
        wk[(size_t)idx * KK + k] = 0.5f * qd1 * mk;
        c1v += qd1 * mk * (0.5f - qdiff);
    }
    c1[idx] = c1v;
    if (t == 0) out[b * SS + (SS - 1)] = 0.0f;   // y[:, S-1] = 0
}

// ===========================================================================
// Kernel 4: sequential scan. One persistent 1024-thread workgroup (32 waves).
// LDS: lp double-buffered (128 KB) + W_forget[0:128,:] staged (64 KB)
//      + contrib (2 KB)  ~= 194 KB, legal within CDNA5's 320 KB WGP LDS.
// Per step: 64 WMMA tiles of lp @ Wg_lds, gate + state update, cosine scoring.
// ===========================================================================
__global__ __launch_bounds__(1024) void k_scan(
    const int* __restrict__ qseq, const int* __restrict__ q2c,
    const float* __restrict__ ec, const float* __restrict__ Wg,
    const float* __restrict__ latent0,
    const float* __restrict__ absorb, const float* __restrict__ fpart,
    const float* __restrict__ wk, const float* __restrict__ c1,
    float* __restrict__ out) {
    __shared__ float lpbuf[2][BB][DD];      // 128 KB (state double buffer)
    __shared__ float wgl[DD][DD];           //  64 KB (W_forget rows 0..127)
    __shared__ float contrib[BB][KK];

    const int tid  = threadIdx.x;
    const int lane = tid & 31;
    const int wv   = tid >> 5;
    const int l16  = lane & 15;
    const int kh   = (lane >> 4) << 1;
    const int mh   = (lane >> 4) << 3;

    for (int i = tid; i < BB * DD; i += 1024) {
        lpbuf[0][i / DD][i % DD] = latent0[i];
        wgl[i / DD][i % DD]      = Wg[i];        // rows 0..127 of W_forget
    }
    __syncthreads();

    for (int t = 0; t < SM1; ++t) {
        const int cur = t & 1, nx = cur ^ 1;
        const float* fpt = fpart  + (size_t)t * BB * DD;
        const float* abt = absorb + (size_t)t * BB * DD;

        // --- forget gate GEMM: lp[128,128] @ Wg[0:128,:] ; 2 tiles / wave ---
        #pragma unroll
        for (int rep = 0; rep < 2; ++rep) {
            const int tile = wv + rep * 32;
            const int mt = tile >> 3, nt = tile & 7;
            const int n  = nt * 16 + l16;
            const int ma = mt * 16 + l16;
            v8f acc = {};
            for (int k0 = 0; k0 < DD; k0 += 4) {
                const int ka = k0 + kh;
                const v2f a = *(const v2f*)(&lpbuf[cur][ma][ka]);  // ds_load_b64
                v2f b;
                b.x = wgl[ka][n];
                b.y = wgl[ka + 1][n];
                acc = wmma_f32_k4(a, b, acc);
            }
            #pragma unroll
            for (int r = 0; r < 8; ++r) {
                const int mm = mt * 16 + r + mh;
                const float fg  = sigmoidf_(acc[r] + fpt[mm * DD + n]);
                lpbuf[nx][mm][n] = lpbuf[cur][mm][n] * fg + abt[mm * DD + n];
            }
        }
        // Prefetch next step's gate slabs into cache while we score this one.
        if (t + 1 < SM1) {
            __builtin_prefetch(fpart  + (size_t)(t + 1) * BB * DD + tid * 16, 0, 0);
            __builtin_prefetch(absorb + (size_t)(t + 1) * BB * DD + tid * 16, 0, 0);
        }
        __syncthreads();

        // --- cosine scoring: thread (b,k) ---
        if (tid < BB * KK) {
            const int b = tid >> 2, k = tid & 3;
            const int q1   = qseq[b * SS + t + 1];
            const int cidx = q2c[q1 * KK + k];
            const float* crow = ec + (size_t)cidx * DD;
            float dot = 0.0f, nl = 0.0f, nc = 0.0f;
            for (int d = 0; d < DD; ++d) {
                const float lv = lpbuf[nx][b][d];
                const float cv = crow[d];
                dot += lv * cv; nl += lv * lv; nc += cv * cv;
            }
            const float cosv = dot / (fmaxf(sqrtf(nl), EPS) * fmaxf(sqrtf(nc), EPS));
            contrib[b][k] = cosv * wk[((size_t)t * BB + b) * KK + k];
        }
        __syncthreads();

        if (tid < BB) {
            const float logit = contrib[tid][0] + contrib[tid][1] +
                                contrib[tid][2] + contrib[tid][3] +
                                c1[(size_t)t * BB + tid];
            out[tid * SS + t] = sigmoidf_(logit);
        }
        __syncthreads();
    }
}

// ===========================================================================
extern "C" void kernel_launch(void* const* d_in, const int* in_sizes, int n_in,
                              void* d_out, int out_size, void* d_ws, size_t ws_size,
                              hipStream_t stream) {
    const int*   question_seq = (const int*)d_in[0];
    const int*   hint_seq     = (const int*)d_in[1];
    const int*   correct_seq  = (const int*)d_in[2];
    const int*   interval_seq = (const int*)d_in[3];
    const int*   use_seq      = (const int*)d_in[4];
    const int*   q2c_table    = (const int*)d_in[5];
    const int*   q2c_mask     = (const int*)d_in[6];
    const float* embed_q      = (const float*)d_in[7];
    const float* embed_qdisc  = (const float*)d_in[8];
    const float* embed_c      = (const float*)d_in[9];
    const float* embed_it     = (const float*)d_in[10];
    const float* embed_ut     = (const float*)d_in[11];
    const float* embed_ha     = (const float*)d_in[12];
    const float* W_fuse       = (const float*)d_in[13];
    const float* b_fuse       = (const float*)d_in[14];
    const float* W_absorb     = (const float*)d_in[15];
    const float* b_absorb     = (const float*)d_in[16];
    const float* W_forget     = (const float*)d_in[17];
    const float* b_forget     = (const float*)d_in[18];
    const float* latent0      = (const float*)d_in[19];
    float* out = (float*)d_out;

    // Workspace carve-up (floats): ~39.6 MB total.
    float* ws = (float*)d_ws;
    const size_t slab = (size_t)SM1 * BB * DD;   // 3,260,416 floats each
    float* inter_ws  = ws;
    float* absorb_ws = ws + slab;
    float* fpart_ws  = ws + 2 * slab;
    float* wk_ws     = ws + 3 * slab;                       // SM1*B*K
    float* c1_ws     = wk_ws + (size_t)SM1 * BB * KK;       // SM1*B

    k_fuse<<<dim3(SM1 * 8), dim3(256), 0, stream>>>(
        question_seq, correct_seq, embed_q, W_fuse, b_fuse, inter_ws);

    k_absorb_forget<<<dim3(SM1 * 8), dim3(256), 0, stream>>>(
        use_seq, hint_seq, interval_seq, inter_ws, embed_ut, embed_ha, embed_it,
        W_absorb, b_absorb, W_forget, b_forget, absorb_ws, fpart_ws);

    k_scalars<<<dim3((SM1 * BB + 255) / 256), dim3(256), 0, stream>>>(
        question_seq, q2c_table, q2c_mask, embed_q, embed_qdisc, embed_c,
        wk_ws, c1_ws, out);

    k_scan<<<dim3(1), dim3(1024), 0, stream>>>(
        question_seq, q2c_table, embed_c, W_forget, latent0,
        absorb_ws, fpart_ws, wk_ws, c1_ws, out);
}